// Upsample_29308856828612
// MI455X (gfx1250) — compile-verified
//
#include <hip/hip_runtime.h>
#include <cstdint>

// upfirdn2d-style 2x upsample, general 4x4 filter read from d_in[1].
// NCHW (16,512,64,64) f32 -> (16,512,127,127) f32.
//
// Memory-bound target: ~632 MB traffic -> ~27us @ 23.3 TB/s.
// CDNA5 path: whole 16KB input plane staged in LDS via
// global_load_async_to_lds_b128 (ASYNCcnt), zero-padded tile so the compute
// loop is branch/select-free: 2 ds_load_b32 + 16 FMA + 4 b32 stores per
// 2x2-output anchor, all with immediate-offset addressing.

#define IH 64
#define IW 64
#define OH 127   // (IH-1)*2 + 1 + (2+1) - 4 + 1
#define OW 127
#define TSTR 72  // padded LDS row stride in floats (interior col c -> c+4, keeps 16B align)
#define TROWS 66 // input rows -1..64 -> tile rows 0..65

__global__ __launch_bounds__(256) void
upsample2x_kernel(const float* __restrict__ x,
                  const float* __restrict__ ker,   // 4x4 effective conv weights
                  float* __restrict__ out)
{
    __shared__ float tile[TROWS * TSTR];   // 19,008 B
    __shared__ float kw[16];

    const int tid   = threadIdx.x;
    const int plane = blockIdx.x;                      // n*C + c
    const float* src = x + (size_t)plane * (IH * IW);
    float* dst       = out + (size_t)plane * (OH * OW);

    if (tid < 16) kw[tid] = ker[tid];

    // ---- Phase 1: async copy plane interior into LDS (gfx1250 ASYNC path) ----
    // input row r -> tile row r+1; input col c -> tile col c+4 (16B aligned).
    const uint32_t lds_base = (uint32_t)(uintptr_t)(&tile[0]);
    {
        const int r0 = tid >> 4;            // 0..15
        const int c0 = (tid & 15) * 4;      // 0..60, multiple of 4
#pragma unroll
        for (int i = 0; i < 4; ++i) {
            const int r = r0 + i * 16;
            const uint64_t gaddr = (uint64_t)(uintptr_t)(src + r * IW + c0);
            const uint32_t laddr = lds_base +
                (uint32_t)(((r + 1) * TSTR + 4 + c0) * sizeof(float));
            asm volatile("global_load_async_to_lds_b128 %0, %1, off"
                         :
                         : "v"(laddr), "v"(gaddr)
                         : "memory");
        }
    }

    // ---- Zero borders (disjoint from async targets, so no ordering needed) ----
    if (tid < TSTR) {
        tile[tid] = 0.0f;                          // tile row 0  (input row -1)
        tile[(TROWS - 1) * TSTR + tid] = 0.0f;     // tile row 65 (input row 64)
    }
    if (tid < IH) {
        tile[(tid + 1) * TSTR + 3] = 0.0f;         // input col -1
    }

    asm volatile("s_wait_asynccnt 0" ::: "memory");
    __syncthreads();

    // Flipped kernel taps in registers: wf[a][b] = ker[3-a][3-b].
    float wf[4][4];
#pragma unroll
    for (int a2 = 0; a2 < 4; ++a2)
#pragma unroll
        for (int b2 = 0; b2 < 4; ++b2)
            wf[a2][b2] = kw[(3 - a2) * 4 + (3 - b2)];

    // Thread -> (column pair, row strip): mp = tid&63 (lane-consecutive for
    // coalescing), row group g = tid>>6 (wave-uniform), 16 anchors vertically.
    // This thread owns output columns (2mp-1, 2mp); needs input cols mp-1, mp.
    const int mp = tid & 63;
    const int g  = tid >> 6;               // 0..3
    const int m0 = g * 16;

    // LDS base for this thread's column pair: tile[.][3+mp] = input col mp-1.
    const float* trow = &tile[3 + mp];

    // Output base pointers (all later addressing is immediate offsets).
    float* rowp  = dst + (size_t)(2 * m0) * OW + (2 * mp - 1);
    // mp==0: left-column output (col -1) doesn't exist; redirect its store to
    // col 0 and overwrite with the correct right-column value (same-lane
    // stores to one address complete in order).
    float* rowpE = rowp + ((mp > 0) ? 0 : 1);

    // Rolling 3x2 window over input rows m-1, m, m+1 at cols mp-1, mp.
    float rv[3][2];
#pragma unroll
    for (int dc = 0; dc < 2; ++dc) {
        rv[0][dc] = trow[(m0)     * TSTR + dc];    // input row m0-1 (tile row m0)
        rv[1][dc] = trow[(m0 + 1) * TSTR + dc];    // input row m0
    }

#pragma unroll
    for (int a = 0; a < 16; ++a) {
        const int m = m0 + a;
        const float* rm1 = rv[a % 3];              // input row m-1
        const float* rm  = rv[(a + 1) % 3];        // input row m
        float*       rp1 = rv[(a + 2) % 3];        // input row m+1 (load now)

        rp1[0] = trow[(m + 2) * TSTR + 0];
        rp1[1] = trow[(m + 2) * TSTR + 1];

        // even output row 2m  : tap rows ky=0 -> m-1, ky=2 -> m
        // odd  output row 2m+1: tap rows ky=1 -> m,   ky=3 -> m+1
        // left col 2mp-1: kx=1 -> mp-1, kx=3 -> mp
        // right col 2mp : kx=0 -> mp-1, kx=2 -> mp
        const float oeL = wf[0][1]*rm1[0] + wf[0][3]*rm1[1]
                        + wf[2][1]*rm[0]  + wf[2][3]*rm[1];
        const float oeR = wf[0][0]*rm1[0] + wf[0][2]*rm1[1]
                        + wf[2][0]*rm[0]  + wf[2][2]*rm[1];
        const float ooL = wf[1][1]*rm[0]  + wf[1][3]*rm[1]
                        + wf[3][1]*rp1[0] + wf[3][3]*rp1[1];
        const float ooR = wf[1][0]*rm[0]  + wf[1][2]*rm[1]
                        + wf[3][0]*rp1[0] + wf[3][2]*rp1[1];

        const int re = (2 * a) * OW;               // compile-time per iteration
        rowpE[re]     = oeL;                       // col 2mp-1 (or dead -> col 0)
        rowp [re + 1] = oeR;                       // col 2mp   (overwrites for mp==0)
        if (a < 15 || g < 3) {                     // bottom output row 127 doesn't exist
            const int ro = (2 * a + 1) * OW;
            rowpE[ro]     = ooL;
            rowp [ro + 1] = ooR;
        }
    }
}

extern "C" void kernel_launch(void* const* d_in, const int* in_sizes, int n_in,
                              void* d_out, int out_size, void* d_ws, size_t ws_size,
                              hipStream_t stream)
{
    const float* x   = (const float*)d_in[0];   // (16,512,64,64) f32
    const float* ker = (const float*)d_in[1];   // (4,4) f32
    float* out       = (float*)d_out;           // (16,512,127,127) f32

    const int planes = in_sizes[0] / (IH * IW); // 8192
    upsample2x_kernel<<<dim3(planes), dim3(256), 0, stream>>>(x, ker, out);
}